// Indexer_15023795602044
// MI455X (gfx1250) — compile-verified
//
#include <hip/hip_runtime.h>
#include <math.h>

// ---------------------------------------------------------------------------
// MI455X (gfx1250) implementation of the hierarchical indexer.
// Gather-bound workload (~1 GB HBM gather at level 2 -> ~46us floor @23.3TB/s).
// Dense GEMMs use exact-fp32 V_WMMA_F32_16X16X4_F32 (precision-safe for top-k).
// ---------------------------------------------------------------------------

typedef __attribute__((ext_vector_type(2))) float v2f;
typedef __attribute__((ext_vector_type(8))) float v8f;

#define INPUT_DIMS 1024
#define IDX_D      128
#define DEPTH      3
#define QCOLS      (DEPTH * IDX_D)   // 384
#define SZ0        512
#define BR1        128
#define BR2        16
#define TOPK       32

__device__ __forceinline__ float sqrt_scale() { return 0.08838834764831845f; } // 1/sqrt(128)

// ---------------------------------------------------------------------------
// Kernel 1: q = X @ Wq + bq   (WMMA f32 16x16x4, one wave per 16x16 tile)
// A 16x4 layout: lanes 0-15 hold M=0..15; v0 = K(0|2), v1 = K(1|3) by lane half.
// C/D: VGPR r -> row (r + 8*half), col = lane%16.
// ---------------------------------------------------------------------------
__global__ void gemm_q_kernel(const float* __restrict__ X,
                              const float* __restrict__ W,
                              const float* __restrict__ bq,
                              float* __restrict__ qout) {
    const int lane = threadIdx.x & 31;
    const int half = lane >> 4;
    const int l    = lane & 15;
    const int m0   = blockIdx.x * 16;
    const int n0   = blockIdx.y * 16;
    const float* xrow = X + (size_t)(m0 + l) * INPUT_DIMS;
    v8f acc = {};
    for (int k = 0; k < INPUT_DIMS; k += 4) {
        const int ka = k + 2 * half;
        v2f a, b;
        a.x = xrow[ka];
        a.y = xrow[ka + 1];
        b.x = W[(size_t)ka * QCOLS + n0 + l];
        b.y = W[(size_t)(ka + 1) * QCOLS + n0 + l];
        acc = __builtin_amdgcn_wmma_f32_16x16x4_f32(false, a, false, b,
                                                    (short)0, acc, false, false);
    }
    const float bias = bq[n0 + l];
#pragma unroll
    for (int r = 0; r < 8; ++r) {
        qout[(size_t)(m0 + r + 8 * half) * QCOLS + n0 + l] = acc[r] + bias;
    }
}

// ---------------------------------------------------------------------------
// Kernel 2: scores = (q[:, :128] @ key0^T) * scale   (WMMA f32 16x16x4)
// B[k][n] = key0[n][k] -> read key0 row-major (it is already N x K).
// ---------------------------------------------------------------------------
__global__ void gemm_scores_kernel(const float* __restrict__ q,
                                   const float* __restrict__ key0,
                                   float* __restrict__ scores) {
    const int lane = threadIdx.x & 31;
    const int half = lane >> 4;
    const int l    = lane & 15;
    const int m0   = blockIdx.x * 16;
    const int n0   = blockIdx.y * 16;
    v8f acc = {};
    for (int k = 0; k < IDX_D; k += 4) {
        const int ka = k + 2 * half;
        v2f a, b;
        a.x = q[(size_t)(m0 + l) * QCOLS + ka];
        a.y = q[(size_t)(m0 + l) * QCOLS + ka + 1];
        b.x = key0[(size_t)(n0 + l) * IDX_D + ka];
        b.y = key0[(size_t)(n0 + l) * IDX_D + ka + 1];
        acc = __builtin_amdgcn_wmma_f32_16x16x4_f32(false, a, false, b,
                                                    (short)0, acc, false, false);
    }
    const float sc = sqrt_scale();
#pragma unroll
    for (int r = 0; r < 8; ++r) {
        scores[(size_t)(m0 + r + 8 * half) * SZ0 + n0 + l] = acc[r] * sc;
    }
}

// ---------------------------------------------------------------------------
// Kernel 3: level-0 top-32 + softmax-renorm.
// softmax -> topk -> renorm == exp(s - s_max)/sum_topk exp(.) (Z cancels).
// Iterative argmax in LDS; tie-break lowest index (matches lax.top_k).
// ---------------------------------------------------------------------------
__global__ void topk0_kernel(const float* __restrict__ scores,
                             float* __restrict__ val_out,
                             int* __restrict__ ind_out) {
    const int row = blockIdx.x;
    const int tid = threadIdx.x;
    const int nth = blockDim.x;   // 256
    __shared__ float s[SZ0];
    __shared__ float rv[256];
    __shared__ int   ri[256];
    __shared__ float tv[TOPK];
    __shared__ int   ti[TOPK];

    for (int c = tid; c < SZ0; c += nth) s[c] = scores[(size_t)row * SZ0 + c];
    __syncthreads();

    for (int j = 0; j < TOPK; ++j) {
        float bv = -3.0e38f;
        int   bi = 0x7fffffff;
        for (int c = tid; c < SZ0; c += nth) {
            const float v = s[c];
            if (v > bv) { bv = v; bi = c; }   // ascending c -> lowest index on tie
        }
        rv[tid] = bv; ri[tid] = bi;
        __syncthreads();
        for (int off = 128; off > 0; off >>= 1) {
            if (tid < off) {
                const float v1 = rv[tid + off];
                const int   i1 = ri[tid + off];
                if (v1 > rv[tid] || (v1 == rv[tid] && i1 < ri[tid])) {
                    rv[tid] = v1; ri[tid] = i1;
                }
            }
            __syncthreads();
        }
        if (tid == 0) { tv[j] = rv[0]; ti[j] = ri[0]; s[ri[0]] = -3.0e38f; }
        __syncthreads();
    }

    if (tid < TOPK) {               // one full wave32
        const float e = expf(tv[tid] - tv[0]);   // tv[0] is the max
        float sum = e;
#pragma unroll
        for (int m = 16; m > 0; m >>= 1) sum += __shfl_xor(sum, m, 32);
        val_out[row * TOPK + tid] = e / sum;
        ind_out[row * TOPK + tid] = ti[tid];
    }
}

// ---------------------------------------------------------------------------
// Kernels 4/5: hierarchical level — gather child keys of previous top-k,
// per-group softmax weighted by parent val, flatten, top-32, renormalize.
// NOTE: reference indexes key_i with the *local flattened* ind (k*branch+b);
// we reproduce that exactly.
// ---------------------------------------------------------------------------
template <int BRANCH>
__global__ void level_topk_kernel(const float* __restrict__ q, int q_col_off,
                                  const float* __restrict__ keys,
                                  const float* __restrict__ val_in,
                                  const int* __restrict__ ind_in,
                                  float* __restrict__ val_out,
                                  int* __restrict__ ind_out) {
    constexpr int N = TOPK * BRANCH;       // 4096 or 512 candidates
    const int row = blockIdx.x;
    const int tid = threadIdx.x;
    const int nth = blockDim.x;            // 256
    __shared__ float qv[IDX_D];
    __shared__ float s[N];
    __shared__ float rv[256];
    __shared__ int   ri[256];
    __shared__ float gv[TOPK];
    __shared__ int   gi[TOPK];
    __shared__ float tv[TOPK];
    __shared__ int   ti[TOPK];

    if (tid < IDX_D) qv[tid] = q[(size_t)row * QCOLS + q_col_off + tid];
    if (tid < TOPK) {
        gv[tid] = val_in[row * TOPK + tid];
        gi[tid] = ind_in[row * TOPK + tid];
    }
    __syncthreads();

    const float sc = sqrt_scale();
    for (int c = tid; c < N; c += nth) {
        const int g = c / BRANCH;
        const int b = c % BRANCH;
        const float* kp = keys + ((size_t)gi[g] * BRANCH + b) * IDX_D;
        // prefetch next gathered key row for this thread (global_prefetch_b8)
        const int cn = c + nth;
        if (cn < N) {
            const float* kpn =
                keys + ((size_t)gi[cn / BRANCH] * BRANCH + (cn % BRANCH)) * IDX_D;
            __builtin_prefetch(kpn, 0, 1);
        }
        float dot = 0.f;
#pragma unroll 8
        for (int d = 0; d < IDX_D; ++d) dot = __builtin_fmaf(qv[d], kp[d], dot);
        s[c] = dot * sc;
    }
    __syncthreads();

    // per-group softmax scaled by parent weight: thread g owns group g
    if (tid < TOPK) {
        const int base = tid * BRANCH;
        float m = -3.0e38f;
        for (int b = 0; b < BRANCH; ++b) m = fmaxf(m, s[base + b]);
        float sum = 0.f;
        for (int b = 0; b < BRANCH; ++b) sum += expf(s[base + b] - m);
        const float wsc = gv[tid] / sum;
        for (int b = 0; b < BRANCH; ++b) s[base + b] = expf(s[base + b] - m) * wsc;
    }
    __syncthreads();

    // iterative top-32 over N non-negative weights
    for (int j = 0; j < TOPK; ++j) {
        float bv = -1.0f;
        int   bi = 0x7fffffff;
        for (int c = tid; c < N; c += nth) {
            const float v = s[c];
            if (v > bv) { bv = v; bi = c; }
        }
        rv[tid] = bv; ri[tid] = bi;
        __syncthreads();
        for (int off = 128; off > 0; off >>= 1) {
            if (tid < off) {
                const float v1 = rv[tid + off];
                const int   i1 = ri[tid + off];
                if (v1 > rv[tid] || (v1 == rv[tid] && i1 < ri[tid])) {
                    rv[tid] = v1; ri[tid] = i1;
                }
            }
            __syncthreads();
        }
        if (tid == 0) { tv[j] = rv[0]; ti[j] = ri[0]; s[ri[0]] = -1.0f; }
        __syncthreads();
    }

    // renormalize selected weights (wave32 butterfly)
    if (tid < TOPK) {
        const float v = tv[tid];
        float sum = v;
#pragma unroll
        for (int m = 16; m > 0; m >>= 1) sum += __shfl_xor(sum, m, 32);
        val_out[row * TOPK + tid] = v / sum;
        ind_out[row * TOPK + tid] = ti[tid];
    }
}

// ---------------------------------------------------------------------------
// Host-side launch. Workspace layout (floats):
//   q       : B*384
//   scores  : B*512
//   val0/ind0/val1/ind1 : B*32 each
// Output: d_out = [ val (B*32 f32) | ind (B*32 i32 bit layout) ]
// ---------------------------------------------------------------------------
extern "C" void kernel_launch(void* const* d_in, const int* in_sizes, int n_in,
                              void* d_out, int out_size, void* d_ws, size_t ws_size,
                              hipStream_t stream) {
    const float* X    = (const float*)d_in[0];
    const float* Wq   = (const float*)d_in[1];
    const float* bq   = (const float*)d_in[2];
    const float* key0 = (const float*)d_in[3];
    const float* key1 = (const float*)d_in[4];
    const float* key2 = (const float*)d_in[5];

    const int B = in_sizes[0] / INPUT_DIMS;   // 4096

    float* ws     = (float*)d_ws;
    float* q      = ws;                                 // B*QCOLS
    float* scores = q + (size_t)B * QCOLS;              // B*SZ0
    float* val0   = scores + (size_t)B * SZ0;           // B*TOPK
    int*   ind0   = (int*)(val0 + (size_t)B * TOPK);    // B*TOPK
    float* val1   = (float*)(ind0 + (size_t)B * TOPK);  // B*TOPK
    int*   ind1   = (int*)(val1 + (size_t)B * TOPK);    // B*TOPK

    float* val_out = (float*)d_out;
    int*   ind_out = (int*)d_out + (size_t)B * TOPK;

    dim3 wave(32);
    gemm_q_kernel<<<dim3(B / 16, QCOLS / 16), wave, 0, stream>>>(X, Wq, bq, q);
    gemm_scores_kernel<<<dim3(B / 16, SZ0 / 16), wave, 0, stream>>>(q, key0, scores);
    topk0_kernel<<<B, 256, 0, stream>>>(scores, val0, ind0);
    level_topk_kernel<BR1><<<B, 256, 0, stream>>>(q, IDX_D, key1, val0, ind0, val1, ind1);
    level_topk_kernel<BR2><<<B, 256, 0, stream>>>(q, 2 * IDX_D, key2, val1, ind1,
                                                  val_out, ind_out);
}